// MoE_72808285602017
// MI455X (gfx1250) — compile-verified
//
#include <hip/hip_runtime.h>
#include <hip/hip_bf16.h>
#include <stdint.h>

// ---------------------------------------------------------------------------
// PropertyLossTracker update for MI455X (gfx1250).
// Bandwidth-bound segment reduction: 128 MB streamed once (~5.5us @ 23.3TB/s).
// - CDNA5 async global->LDS DMA (ASYNCcnt) with double buffering
// - LDS-privatized histograms (ds_add_f32 / ds_add_u32)
// - store+tree-reduce flush (avoids ~8M serialized global atomics)
// ---------------------------------------------------------------------------

#define P_SEG     4096
#define BLOCK     256
#define NWAVES    (BLOCK / 32)      // wave32: 8 waves per block
#define TILE      128               // tokens per wave per async tile
#define MAX_PARTS 512

// Issue one per-lane 16B async copy global -> LDS (GVS addressing mode:
// 64-bit SGPR base + 32-bit VGPR byte offset). Tracked by ASYNCcnt.
__device__ __forceinline__ void async_copy_b128(unsigned lds_byte_off,
                                                unsigned global_byte_off,
                                                const void* base) {
  asm volatile("global_load_async_to_lds_b128 %0, %1, %2"
               :: "v"(lds_byte_off), "v"(global_byte_off), "s"(base)
               : "memory");
}

// Low 32 bits of a flat shared-aperture pointer == LDS byte offset
// (ISA 10.2).
__device__ __forceinline__ unsigned lds_off(const void* p) {
  return (unsigned)(uintptr_t)p;
}

__global__ void __launch_bounds__(BLOCK)
hist_kernel(const int* __restrict__ ids,
            const float* __restrict__ losses,
            float* __restrict__ g_sums,   // [P_SEG]   (atomic-fallback mode)
            float* __restrict__ g_cnts,   // [P_SEG]   (atomic-fallback mode)
            float* __restrict__ part,     // [G][2*P_SEG] (partials mode)
            int use_part,
            int n) {
  __shared__ float    h_sum[P_SEG];
  __shared__ unsigned h_cnt[P_SEG];
  // [wave][buf][0=ids,1=losses][TILE] staging for async DMA (16 KB)
  __shared__ __align__(16) unsigned stage[NWAVES][2][2][TILE];

  for (int i = threadIdx.x; i < P_SEG; i += BLOCK) {
    h_sum[i] = 0.0f;
    h_cnt[i] = 0u;
  }
  __syncthreads();

  const int wave = threadIdx.x >> 5;
  const int lane = threadIdx.x & 31;
  const long long gwave  = (long long)blockIdx.x * NWAVES + wave;
  const long long nwaves = (long long)gridDim.x * NWAVES;
  const long long ntiles = (long long)n / TILE;

  // --- double-buffered async pipeline over full tiles (wave-uniform ctrl) ---
  long long t = gwave;
  int buf = 0;
  if (t < ntiles) {
    const unsigned goff = (unsigned)((t * TILE + lane * 4) * 4);
    async_copy_b128(lds_off(&stage[wave][0][0][lane * 4]), goff, ids);
    async_copy_b128(lds_off(&stage[wave][0][1][lane * 4]), goff, losses);
  }
  for (; t < ntiles; t += nwaves) {
    const long long tn = t + nwaves;
    if (tn < ntiles) {
      const unsigned goff = (unsigned)((tn * TILE + lane * 4) * 4);
      async_copy_b128(lds_off(&stage[wave][buf ^ 1][0][lane * 4]), goff, ids);
      async_copy_b128(lds_off(&stage[wave][buf ^ 1][1][lane * 4]), goff, losses);
      // the 2 just-issued may stay outstanding; current buffer is complete
      asm volatile("s_wait_asynccnt 0x2" ::: "memory");
    } else {
      asm volatile("s_wait_asynccnt 0x0" ::: "memory");
    }

    const uint4  iv = *reinterpret_cast<const uint4*>(&stage[wave][buf][0][lane * 4]);
    const float4 lv = *reinterpret_cast<const float4*>(&stage[wave][buf][1][lane * 4]);

    unsigned i0 = iv.x & (P_SEG - 1);
    unsigned i1 = iv.y & (P_SEG - 1);
    unsigned i2 = iv.z & (P_SEG - 1);
    unsigned i3 = iv.w & (P_SEG - 1);
    atomicAdd(&h_sum[i0], lv.x); atomicAdd(&h_cnt[i0], 1u);
    atomicAdd(&h_sum[i1], lv.y); atomicAdd(&h_cnt[i1], 1u);
    atomicAdd(&h_sum[i2], lv.z); atomicAdd(&h_cnt[i2], 1u);
    atomicAdd(&h_sum[i3], lv.w); atomicAdd(&h_cnt[i3], 1u);

    buf ^= 1;
  }

  // --- remainder tokens (none for N = 16M, but stay general) ---
  for (long long i = ntiles * TILE + (long long)blockIdx.x * BLOCK + threadIdx.x;
       i < (long long)n;
       i += (long long)gridDim.x * BLOCK) {
    unsigned id = ((unsigned)ids[i]) & (P_SEG - 1);
    atomicAdd(&h_sum[id], losses[i]);
    atomicAdd(&h_cnt[id], 1u);
  }

  __syncthreads();

  if (use_part) {
    // Coalesced b128 store of this block's private histogram to its slice.
    float* p_sum = part + (size_t)blockIdx.x * (2 * P_SEG);
    float* p_cnt = p_sum + P_SEG;
    const int b0 = threadIdx.x * 16;            // 16 bins per thread
    #pragma unroll
    for (int k = 0; k < 16; k += 4) {
      const int b = b0 + k;
      const float4 s4 = *reinterpret_cast<const float4*>(&h_sum[b]);
      const uint4  c4 = *reinterpret_cast<const uint4*>(&h_cnt[b]);
      float4 cf;
      cf.x = (float)c4.x; cf.y = (float)c4.y;
      cf.z = (float)c4.z; cf.w = (float)c4.w;
      *reinterpret_cast<float4*>(&p_sum[b]) = s4;
      *reinterpret_cast<float4*>(&p_cnt[b]) = cf;
    }
  } else {
    // Fallback: global-atomic flush (small scratch).
    for (int i = threadIdx.x; i < P_SEG; i += BLOCK) {
      unsigned c = h_cnt[i];
      if (c != 0u) {
        atomicAdd(&g_sums[i], h_sum[i]);
        atomicAdd(&g_cnts[i], (float)c);
      }
    }
  }
}

// Reduce G per-block partial histograms into the final [P_SEG] arrays.
// Lane-consecutive bins -> fully coalesced reads at every g.
__global__ void __launch_bounds__(256)
reduce_kernel(const float* __restrict__ part,
              float* __restrict__ final_sums,
              float* __restrict__ final_cnts,
              int G) {
  const int j = blockIdx.x * 256 + threadIdx.x;     // 0 .. 2*P_SEG-1
  const int is_cnt = j >= P_SEG;
  const int bin = is_cnt ? (j - P_SEG) : j;
  const float* base = part + (is_cnt ? P_SEG : 0) + bin;
  float acc = 0.0f;
  for (int g = 0; g < G; ++g) {
    acc += base[(size_t)g * (2 * P_SEG)];
  }
  if (is_cnt) final_cnts[bin] = acc;
  else        final_sums[bin] = acc;
}

__global__ void __launch_bounds__(1024)
finalize_kernel(const float* __restrict__ g_sums,
                const float* __restrict__ g_cnts,
                const float* __restrict__ prop_freq,
                const int* __restrict__ bc_ptr,
                float* __restrict__ out,   // [0]=stratified [1]=unweighted [2..]=new_freq
                int n) {
  __shared__ float r1[1024];   // sum of raw weights (present)
  __shared__ float r2[1024];   // sum of mean_loss * raw
  __shared__ float r3[1024];   // total token-loss sum

  const int tid = threadIdx.x;
  const int bc  = *bc_ptr;
  const float nf32 = (float)n;

  float s1 = 0.0f, s2 = 0.0f, s3 = 0.0f;
  for (int p = tid; p < P_SEG; p += 1024) {
    const float c = g_cnts[p];
    const float s = g_sums[p];
    const bool present = c > 0.0f;

    const float mean  = present ? (s / fmaxf(c, 1.0f)) : 0.0f;
    const float bfreq = c / (nf32 + 1e-6f);
    const float nfreq = prop_freq[p] * 0.99f + (present ? 0.01f * bfreq : 0.0f);
    out[2 + p] = nfreq;

    const float fcl = fmaxf(nfreq, 1e-5f);
    float raw = 1.0f / sqrtf(fcl + 1e-6f);          // POWER = 0.5
    const float ramp = fminf(1.0f, ((float)bc - 1000.0f) * (1.0f / 200.0f));
    raw = 1.0f + ramp * (raw - 1.0f);
    raw = fminf(30.0f, raw);
    if (bc <= 3000) {
      const float frac = (float)bc / 3000.0f;
      raw = raw * frac + (1.0f - frac);
    }
    if (bc <= 1000) raw = 1.0f;

    const float w = present ? raw : 0.0f;
    s1 += w;
    s2 += mean * w;
    s3 += s;
  }

  r1[tid] = s1; r2[tid] = s2; r3[tid] = s3;
  __syncthreads();
  for (int off = 512; off > 0; off >>= 1) {
    if (tid < off) {
      r1[tid] += r1[tid + off];
      r2[tid] += r2[tid + off];
      r3[tid] += r3[tid + off];
    }
    __syncthreads();
  }
  if (tid == 0) {
    out[0] = r2[0] / (r1[0] + 1e-6f);   // stratified loss
    out[1] = r3[0] / nf32;              // unweighted mean
  }
}

__global__ void zero_kernel(float* ws, int n) {
  const int i = blockIdx.x * 256 + threadIdx.x;
  if (i < n) ws[i] = 0.0f;
}

extern "C" void kernel_launch(void* const* d_in, const int* in_sizes, int n_in,
                              void* d_out, int out_size, void* d_ws, size_t ws_size,
                              hipStream_t stream) {
  const int*   property_ids = (const int*)d_in[0];
  const float* token_losses = (const float*)d_in[1];
  const float* prop_freq    = (const float*)d_in[2];
  const int*   batch_cnt    = (const int*)d_in[3];
  float*       out          = (float*)d_out;
  const int    n            = in_sizes[0];

  float* final_sums = (float*)d_ws;
  float* final_cnts = final_sums + P_SEG;
  float* part       = final_cnts + P_SEG;

  const size_t per_part = (size_t)(2 * P_SEG) * sizeof(float);   // 32 KB
  long long cap = ((long long)ws_size - (long long)per_part) / (long long)per_part;
  if (cap < 0) cap = 0;
  int G = (int)(cap < MAX_PARTS ? cap : MAX_PARTS);

  if (G >= 64) {
    // Store + tree-reduce flush: ~16 MB extra traffic instead of ~8M atomics.
    hist_kernel<<<G, BLOCK, 0, stream>>>(property_ids, token_losses,
                                         final_sums, final_cnts, part,
                                         /*use_part=*/1, n);
    reduce_kernel<<<(2 * P_SEG) / 256, 256, 0, stream>>>(part, final_sums,
                                                         final_cnts, G);
  } else {
    // Small-scratch fallback: atomic flush with modest grid.
    zero_kernel<<<(2 * P_SEG + 255) / 256, 256, 0, stream>>>(final_sums,
                                                             2 * P_SEG);
    hist_kernel<<<256, BLOCK, 0, stream>>>(property_ids, token_losses,
                                           final_sums, final_cnts, part,
                                           /*use_part=*/0, n);
  }

  finalize_kernel<<<1, 1024, 0, stream>>>(final_sums, final_cnts, prop_freq,
                                          batch_cnt, out, n);
}